// ConcatNonLinear_180388626641
// MI455X (gfx1250) — compile-verified
//
#include <hip/hip_runtime.h>
#include <hip/hip_bf16.h>

#define N_U 100000
#define N_V 50000
#define N_TOT (N_U + N_V)
#define K_EMB 64
#define COMB 256              // K*(L+1)

typedef __attribute__((ext_vector_type(2))) float v2f;
typedef __attribute__((ext_vector_type(8))) float v8f;

// ---------------------------------------------------------------------------
// Kernel 1: initialize E_comb: col block 0 = [E_u; E_v], blocks 1..3 = 0.
// Vectorized: one float4 (b128) store per thread. Must run every launch
// (the SpMM atomics accumulate into blocks 1..3).
// ---------------------------------------------------------------------------
__global__ void init_ecomb_kernel(const float* __restrict__ Eu,
                                  const float* __restrict__ Ev,
                                  float* __restrict__ E) {
    long gid = (long)blockIdx.x * blockDim.x + threadIdx.x;   // over N_TOT*64
    if (gid >= (long)N_TOT * K_EMB) return;
    int row   = (int)(gid >> 6);
    int chunk = (int)(gid & 63);          // 4-float chunk within the 256 cols
    float4 v = make_float4(0.f, 0.f, 0.f, 0.f);
    if (chunk < 16) {                     // cols 0..63 come from E0
        const float* src = (row < N_U) ? (Eu + (size_t)row * K_EMB)
                                       : (Ev + (size_t)(row - N_U) * K_EMB);
        v = *(const float4*)(src + chunk * 4);
    }
    *(float4*)(E + (size_t)row * COMB + chunk * 4) = v;
}

// ---------------------------------------------------------------------------
// Kernel 2: one SpMM layer. 16 lanes per nonzero edge; each lane handles
// four consecutive columns (b128 gather, four global_atomic_add_f32 scatters).
// Working set (input slice 38.4 MB, output slice 38.4 MB) fits in the 192 MB
// L2, so this is L2-request bound -> minimize instruction count.
// ---------------------------------------------------------------------------
__global__ void spmm_layer_kernel(const int* __restrict__ Arow,
                                  const int* __restrict__ Acol,
                                  const float* __restrict__ Aval,
                                  float* __restrict__ E,
                                  int nnz, int in_off, int out_off) {
    long gid = (long)blockIdx.x * blockDim.x + threadIdx.x;
    int e    = (int)(gid >> 4);
    if (e >= nnz) return;
    int lane = (int)(gid & 15);

    int   src = Acol[e];
    int   dst = Arow[e];
    float w   = Aval[e];

    const float* sp = E + (size_t)src * COMB + in_off + lane * 4;
    float4 d = *(const float4*)sp;

    float* dp = E + (size_t)dst * COMB + out_off + lane * 4;
    atomicAdd(dp + 0, w * d.x);
    atomicAdd(dp + 1, w * d.y);
    atomicAdd(dp + 2, w * d.z);
    atomicAdd(dp + 3, w * d.w);
}

// ---------------------------------------------------------------------------
// Kernel 3: fused gather + MLP using V_WMMA_F32_16X16X4_F32.
// One wave per 16 batch rows. Accumulates 4 tiles of 16x16 (N = 64 total),
// K = 512 (user 256 + item 256), then bias + exact GELU + dot with W2,
// 16-lane butterfly reduction, write 16 outputs.
// ---------------------------------------------------------------------------
__device__ __forceinline__ float gelu_exact(float x) {
    return 0.5f * x * (1.0f + erff(x * 0.70710678118654752440f));
}

__global__ void mlp_wmma_kernel(const int* __restrict__ users,
                                const int* __restrict__ items,
                                const float* __restrict__ E,
                                const float* __restrict__ W1,   // [512,64] row-major
                                const float* __restrict__ b1,   // [64]
                                const float* __restrict__ W2,   // [64]
                                const float* __restrict__ b2,   // [1]
                                float* __restrict__ out,        // [B]
                                int B) {
    int tid  = blockIdx.x * blockDim.x + threadIdx.x;
    int wave = tid >> 5;
    int lane = tid & 31;
    int m    = lane & 15;     // A: row-in-tile (lanes 0-15/16-31 both map M=0..15)
    int half = lane >> 4;     // selects K-pair {0,1} vs {2,3} for A and B
    int rowb = wave * 16;
    if (rowb >= B) return;

    int b_m  = rowb + m;
    const float* xu = E + (size_t)users[b_m] * COMB;
    const float* xi = E + (size_t)(N_U + items[b_m]) * COMB;

    v8f acc0 = {}, acc1 = {}, acc2 = {}, acc3 = {};

    #pragma unroll
    for (int part = 0; part < 2; ++part) {
        const float* xa    = part ? xi : xu;
        const float* wbase = W1 + (size_t)part * 256 * 64;
        for (int kk = 0; kk < 256; kk += 4) {
            int kr = kk + 2 * half;
            // A fragment: lane (half,m) -> x[row m][kr], x[row m][kr+1]
            v2f a;
            a[0] = xa[kr];
            a[1] = xa[kr + 1];
            // B fragments: lane (half,m) -> W1[kr][n], W1[kr+1][n], n = t*16+m
            const float* w = wbase + (size_t)kr * 64 + m;
            v2f bf0; bf0[0] = w[0];   bf0[1] = w[64];
            v2f bf1; bf1[0] = w[16];  bf1[1] = w[80];
            v2f bf2; bf2[0] = w[32];  bf2[1] = w[96];
            v2f bf3; bf3[0] = w[48];  bf3[1] = w[112];

            acc0 = __builtin_amdgcn_wmma_f32_16x16x4_f32(false, a, false, bf0,
                                                         (short)0, acc0, false, false);
            acc1 = __builtin_amdgcn_wmma_f32_16x16x4_f32(false, a, false, bf1,
                                                         (short)0, acc1, false, false);
            acc2 = __builtin_amdgcn_wmma_f32_16x16x4_f32(false, a, false, bf2,
                                                         (short)0, acc2, false, false);
            acc3 = __builtin_amdgcn_wmma_f32_16x16x4_f32(false, a, false, bf3,
                                                         (short)0, acc3, false, false);
        }
    }

    // Epilogue: h[M][n] = gelu(acc_t[r] + b1[n]), n = t*16+m, M = r + 8*half.
    // out[M] = sum_n h[M][n]*W2[n] + b2  -> reduce n over 4 tiles (in-lane)
    // and over the 16 lanes of this half (xor butterfly).
    float w2v0 = W2[m],      w2v1 = W2[16 + m];
    float w2v2 = W2[32 + m], w2v3 = W2[48 + m];
    float bb0  = b1[m],      bb1  = b1[16 + m];
    float bb2  = b1[32 + m], bb3  = b1[48 + m];

    float red[8];
    #pragma unroll
    for (int r = 0; r < 8; ++r) {
        float s = gelu_exact(acc0[r] + bb0) * w2v0
                + gelu_exact(acc1[r] + bb1) * w2v1
                + gelu_exact(acc2[r] + bb2) * w2v2
                + gelu_exact(acc3[r] + bb3) * w2v3;
        #pragma unroll
        for (int off = 1; off < 16; off <<= 1)
            s += __shfl_xor(s, off, 32);     // stays within the 16-lane half
        red[r] = s;
    }
    if (m < 8)
        out[rowb + 8 * half + m] = red[m] + b2[0];
}

// ---------------------------------------------------------------------------
// Launch: init -> 3x SpMM -> fused WMMA MLP. All on `stream`.
// Input order: users, items, A_rows, A_cols, A_vals, E_u, E_v, W1, b1, W2, b2
// ---------------------------------------------------------------------------
extern "C" void kernel_launch(void* const* d_in, const int* in_sizes, int n_in,
                              void* d_out, int out_size, void* d_ws, size_t ws_size,
                              hipStream_t stream) {
    const int*   users = (const int*)  d_in[0];
    const int*   items = (const int*)  d_in[1];
    const int*   Arow  = (const int*)  d_in[2];
    const int*   Acol  = (const int*)  d_in[3];
    const float* Aval  = (const float*)d_in[4];
    const float* Eu    = (const float*)d_in[5];
    const float* Ev    = (const float*)d_in[6];
    const float* W1    = (const float*)d_in[7];
    const float* b1    = (const float*)d_in[8];
    const float* W2    = (const float*)d_in[9];
    const float* b2    = (const float*)d_in[10];
    float*       out   = (float*)d_out;

    int nnz = in_sizes[2];
    int B   = in_sizes[0];

    float* Ecomb = (float*)d_ws;   // [N_TOT, 256] = ~153.6 MB

    // 1) init E_comb (E0 + zero accumulation blocks) — every launch.
    {
        long total = (long)N_TOT * K_EMB;   // one float4 chunk per thread
        int blocks = (int)((total + 255) / 256);
        init_ecomb_kernel<<<blocks, 256, 0, stream>>>(Eu, Ev, Ecomb);
    }
    // 2) three SpMM layers (col block l -> l+1), 16 lanes per edge.
    {
        long threads = (long)nnz * 16;
        int blocks = (int)((threads + 255) / 256);
        spmm_layer_kernel<<<blocks, 256, 0, stream>>>(Arow, Acol, Aval, Ecomb,
                                                      nnz, 0, 64);
        spmm_layer_kernel<<<blocks, 256, 0, stream>>>(Arow, Acol, Aval, Ecomb,
                                                      nnz, 64, 128);
        spmm_layer_kernel<<<blocks, 256, 0, stream>>>(Arow, Acol, Aval, Ecomb,
                                                      nnz, 128, 192);
    }
    // 3) fused gather + MLP (WMMA f32).
    {
        int waves  = B / 16;            // B = 16384 -> 1024 waves
        int blocks = (waves * 32 + 255) / 256;
        mlp_wmma_kernel<<<blocks, 256, 0, stream>>>(users, items, Ecomb,
                                                    W1, b1, W2, b2, out, B);
    }
}